// DecoderGRUCoverFixed_7026566496793
// MI455X (gfx1250) — compile-verified
//
#include <hip/hip_runtime.h>
#include <math.h>

typedef float v2f __attribute__((ext_vector_type(2)));
typedef float v8f __attribute__((ext_vector_type(8)));

#define HDIM 256
#define VDIM 5000
#define CDIM 32
#define ADM  50
#define HOPS 20

#define BM 128
#define BN 64
#define BK 32

#if defined(__HIP_DEVICE_COMPILE__) && __has_builtin(__builtin_amdgcn_tensor_load_to_lds) && __has_builtin(__builtin_amdgcn_s_wait_tensorcnt)
#define USE_TDM 1
#else
#define USE_TDM 0
#endif

#if USE_TDM
typedef unsigned int u32x4 __attribute__((ext_vector_type(4)));
typedef int i32x4 __attribute__((ext_vector_type(4)));
typedef int i32x8 __attribute__((ext_vector_type(8)));

// Truncate a generic pointer into LDS to its 32-bit LDS byte address
// (flat LDS aperture: the low 32 bits are the LDS address).
__device__ __forceinline__ unsigned lds_addr_of(const void* p) {
    return (unsigned)(unsigned long long)(size_t)p;
}

// 2D TDM tile load, fp32 elements, with LDS row padding of 1 DWORD every
// (1<<pad_code)*2 DWORDs. OOB reads (x>=tensor_w or y>=tensor_h) return zero.
__device__ __forceinline__ void tdm_load_tile_f32(
    const float* gptr, unsigned lds_byte_addr,
    unsigned tensor_w, unsigned tensor_h,
    unsigned tile_w, unsigned tile_h,
    unsigned long long row_stride_elems, unsigned pad_code)
{
    const unsigned long long ga = (unsigned long long)(size_t)(const void*)gptr;

    u32x4 g0;
    g0.x = 1u;                                                  // count=1 (valid user D#)
    g0.y = lds_byte_addr;                                       // lds_addr [63:32]
    g0.z = (unsigned)(ga & 0xffffffffull);                      // global_addr[31:0]
    g0.w = (unsigned)((ga >> 32) & 0x1ffffffull) | (2u << 30);  // global_addr[56:32] | type=2

    i32x8 g1;
    // data_size=2 (4B) | pad_enable | pad_interval=pad_code | pad_amount=0 (1 DWORD)
    g1[0] = (int)((2u << 16) | (1u << 20) | ((pad_code & 7u) << 22));
    g1[1] = (int)((tensor_w & 0xffffu) << 16);                  // atomic_addr=0 | tensor_dim0[15:0]
    g1[2] = (int)(((tensor_w >> 16) & 0xffffu) | ((tensor_h & 0xffffu) << 16));
    g1[3] = (int)(((tensor_h >> 16) & 0xffffu) | ((tile_w & 0xffffu) << 16));
    g1[4] = (int)(tile_h & 0xffffu);                            // tile_dim1 | tile_dim2=0
    g1[5] = (int)(unsigned)(row_stride_elems & 0xffffffffull);  // tensor_dim0_stride[31:0]
    g1[6] = (int)(unsigned)((row_stride_elems >> 32) & 0xffffull); // stride[47:32] | dim1_stride lo=0
    g1[7] = 0;

    i32x4 zz4; zz4[0] = 0; zz4[1] = 0; zz4[2] = 0; zz4[3] = 0;
    i32x8 zz8; zz8[0] = 0; zz8[1] = 0; zz8[2] = 0; zz8[3] = 0;
               zz8[4] = 0; zz8[5] = 0; zz8[6] = 0; zz8[7] = 0;
    // 6-arg toolchain variant: (g0, g1, g2, g3, extra, cpol)
    __builtin_amdgcn_tensor_load_to_lds(g0, g1, zz4, zz4, zz8, 0);
}
#endif  // USE_TDM

// ---------------- Generic WMMA f32 GEMM ----------------
// C[M,N] = act(A[M,K] @ B[K,N] + bias), row-major, act: 0=none 1=relu
__global__ __launch_bounds__(256)
void gemm_f32_wmma(const float* __restrict__ A, const float* __restrict__ B,
                   const float* __restrict__ bias, float* __restrict__ Cm,
                   int M, int N, int K, int act)
{
    __shared__ float sA[2][BM][BK + 1];
    __shared__ float sB[2][BK][BN + 1];

    const int tid    = threadIdx.x;
    const int wave   = tid >> 5;
    const int lane   = tid & 31;
    const int lane16 = lane & 15;
    const int khalf  = (lane >> 4) << 1;     // 0 (lanes 0-15) or 2 (lanes 16-31)

    const int wm = wave >> 1;                // 0..3 -> 32-row strip
    const int wn = wave & 1;                 // 0..1 -> 32-col strip

    const int blockM = blockIdx.y * BM;
    const int blockN = blockIdx.x * BN;

    const v8f vzero = {0.f,0.f,0.f,0.f,0.f,0.f,0.f,0.f};
    v8f acc[2][2];
    acc[0][0] = vzero; acc[0][1] = vzero; acc[1][0] = vzero; acc[1][1] = vzero;

    int cur = 0;

#if USE_TDM
    // Prologue: DMA stage 0 into buffer 0 (wave 0 issues; TDM ignores EXEC).
    if (wave == 0) {
        tdm_load_tile_f32(A + (size_t)blockM * (size_t)K, lds_addr_of(&sA[0][0][0]),
                          (unsigned)K, (unsigned)(M - blockM), BK, BM,
                          (unsigned long long)K, /*32 DW rows*/ 4u);
        tdm_load_tile_f32(B + blockN, lds_addr_of(&sB[0][0][0]),
                          (unsigned)(N - blockN), (unsigned)K, BN, BK,
                          (unsigned long long)N, /*64 DW rows*/ 5u);
        __builtin_amdgcn_s_wait_tensorcnt(0);
    }
    __syncthreads();
#endif

    for (int k0 = 0; k0 < K; k0 += BK) {
#if USE_TDM
        // Kick off the next stage's DMA into the alternate buffer.
        const int nk = k0 + BK;
        if (nk < K && wave == 0) {
            tdm_load_tile_f32(A + (size_t)blockM * (size_t)K + nk,
                              lds_addr_of(&sA[cur ^ 1][0][0]),
                              (unsigned)(K - nk), (unsigned)(M - blockM), BK, BM,
                              (unsigned long long)K, 4u);
            tdm_load_tile_f32(B + (size_t)nk * (size_t)N + blockN,
                              lds_addr_of(&sB[cur ^ 1][0][0]),
                              (unsigned)(N - blockN), (unsigned)(K - nk), BN, BK,
                              (unsigned long long)N, 5u);
        }
#else
        // ---- synchronous fallback staging into buffer 0 ----
        {
            const int col = (tid & 7) * 4;
            #pragma unroll
            for (int r = 0; r < 4; ++r) {
                const int row = (tid >> 3) + r * 32;
                const int gr  = blockM + row;
                const int gc  = k0 + col;
                float4 val = make_float4(0.f, 0.f, 0.f, 0.f);
                if (gr < M) {
                    const float* ap = A + (size_t)gr * (size_t)K + gc;
                    if (gc + 3 < K) {
                        val = *(const float4*)ap;
                    } else {
                        if (gc     < K) val.x = ap[0];
                        if (gc + 1 < K) val.y = ap[1];
                        if (gc + 2 < K) val.z = ap[2];
                        if (gc + 3 < K) val.w = ap[3];
                    }
                    if (gc + BK < K) __builtin_prefetch(ap + BK, 0, 1);
                }
                sA[0][row][col]     = val.x;
                sA[0][row][col + 1] = val.y;
                sA[0][row][col + 2] = val.z;
                sA[0][row][col + 3] = val.w;
            }
        }
        {
            const int col = (tid & 15) * 4;
            #pragma unroll
            for (int r = 0; r < 2; ++r) {
                const int row = (tid >> 4) + r * 16;
                const int gr  = k0 + row;
                const int gc  = blockN + col;
                float4 val = make_float4(0.f, 0.f, 0.f, 0.f);
                if (gr < K) {
                    const float* bp = B + (size_t)gr * (size_t)N + gc;
                    if (gc + 3 < N) {
                        val = *(const float4*)bp;
                    } else {
                        if (gc     < N) val.x = bp[0];
                        if (gc + 1 < N) val.y = bp[1];
                        if (gc + 2 < N) val.z = bp[2];
                        if (gc + 3 < N) val.w = bp[3];
                    }
                }
                sB[0][row][col]     = val.x;
                sB[0][row][col + 1] = val.y;
                sB[0][row][col + 2] = val.z;
                sB[0][row][col + 3] = val.w;
            }
        }
        __syncthreads();
#endif

        // ---- 8 WMMA k-steps over the BK=32 stage ----
        #pragma unroll
        for (int ks = 0; ks < BK; ks += 4) {
            v2f a[2], b[2];
            #pragma unroll
            for (int i = 0; i < 2; ++i) {
                const int mm = wm * 32 + i * 16 + lane16;
                a[i].x = sA[cur][mm][ks + khalf];
                a[i].y = sA[cur][mm][ks + khalf + 1];
            }
            #pragma unroll
            for (int j = 0; j < 2; ++j) {
                const int nn = wn * 32 + j * 16 + lane16;
                b[j].x = sB[cur][ks + khalf][nn];
                b[j].y = sB[cur][ks + khalf + 1][nn];
            }
            #pragma unroll
            for (int i = 0; i < 2; ++i)
                #pragma unroll
                for (int j = 0; j < 2; ++j)
                    acc[i][j] = __builtin_amdgcn_wmma_f32_16x16x4_f32(
                        false, a[i], false, b[j], (short)0, acc[i][j], false, false);
        }

#if USE_TDM
        if (wave == 0) __builtin_amdgcn_s_wait_tensorcnt(0);
        __syncthreads();
        cur ^= 1;
#else
        __syncthreads();
#endif
    }

    // ---- store (C/D layout: VGPR r -> M=r lanes 0-15, M=r+8 lanes 16-31) ----
    #pragma unroll
    for (int i = 0; i < 2; ++i) {
        const int rbase = blockM + wm * 32 + i * 16 + ((lane >> 4) << 3);
        #pragma unroll
        for (int j = 0; j < 2; ++j) {
            const int col = blockN + wn * 32 + j * 16 + lane16;
            if (col < N) {
                const float bv = bias ? bias[col] : 0.f;
                #pragma unroll
                for (int r = 0; r < 8; ++r) {
                    const int row = rbase + r;
                    if (row < M) {
                        float v = acc[i][j][r] + bv;
                        if (act == 1) v = fmaxf(v, 0.f);
                        Cm[(size_t)row * (size_t)N + col] = v;
                    }
                }
            }
        }
    }
}

// ---------------- 20-hop coverage scan (single workgroup) ----------------
__global__ __launch_bounds__(256)
void hop_kernel(const float* __restrict__ query, const float* __restrict__ keys,
                const float* __restrict__ Wattn, const float* __restrict__ battn,
                const float* __restrict__ v_cov,
                const float* __restrict__ Wcov, const float* __restrict__ bcov,
                const float* __restrict__ Watt, const float* __restrict__ batt,
                const float* __restrict__ Wde,  const float* __restrict__ bde,
                const float* __restrict__ Wrg,  const float* __restrict__ brg,
                const float* __restrict__ Kw,   const float* __restrict__ Ge,
                const float* __restrict__ Mve,
                float* __restrict__ q_out, float* __restrict__ cov_out)
{
    __shared__ float q[HDIM], qW[HDIM], rd[HDIM], tmp[HDIM];
    __shared__ float cov[ADM * CDIM], covN[ADM * CDIM];
    __shared__ float w[ADM], sc[ADM];
    __shared__ float red[256];
    __shared__ float gd[3 * CDIM];

    const int t = threadIdx.x;
    q[t] = query[t];
    for (int i = t; i < ADM * CDIM; i += 256) cov[i] = 0.f;
    __syncthreads();

    for (int hop = 0; hop < HOPS; ++hop) {
        // qW = q @ Wattn[0:H] + battn
        float aq = battn[t];
        for (int i = 0; i < HDIM; ++i) aq += q[i] * Wattn[i * HDIM + t];
        qW[t] = aq;
        __syncthreads();

        // additive attention scores over ADM memories
        const float* Wc = Wattn + (size_t)(2 * HDIM) * HDIM;   // cov rows 512..543
        for (int a = 0; a < ADM; ++a) {
            float cvw = 0.f;
            for (int c = 0; c < CDIM; ++c) cvw += cov[a * CDIM + c] * Wc[c * HDIM + t];
            const float e = tanhf(qW[t] + Kw[a * HDIM + t] + cvw);
            red[t] = e * v_cov[t];
            __syncthreads();
            for (int s = 128; s > 0; s >>= 1) {
                if (t < s) red[t] += red[t + s];
                __syncthreads();
            }
            if (t == 0) sc[a] = red[0];
            __syncthreads();
        }
        if (t == 0) {
            float mx = sc[0];
            for (int a = 1; a < ADM; ++a) mx = fmaxf(mx, sc[a]);
            float s = 0.f;
            for (int a = 0; a < ADM; ++a) { w[a] = __expf(sc[a] - mx); s += w[a]; }
            const float inv = 1.f / s;
            for (int a = 0; a < ADM; ++a) w[a] *= inv;
        }
        __syncthreads();

        // gd = q @ Wde + bde
        if (t < 3 * CDIM) {
            float a2 = bde[t];
            for (int i = 0; i < HDIM; ++i) a2 += q[i] * Wde[i * (3 * CDIM) + t];
            gd[t] = a2;
        }
        __syncthreads();

        // CoverageGRUCell update
        for (int idx = t; idx < ADM * CDIM; idx += 256) {
            const int a = idx / CDIM, c = idx % CDIM;
            float cz = bcov[c], cr = bcov[CDIM + c], cn = bcov[2 * CDIM + c];
            for (int cc = 0; cc < CDIM; ++cc) {
                const float cv = cov[a * CDIM + cc];
                cz += cv * Wcov[cc * (3 * CDIM) + c];
                cr += cv * Wcov[cc * (3 * CDIM) + CDIM + c];
                cn += cv * Wcov[cc * (3 * CDIM) + 2 * CDIM + c];
            }
            const float wa = w[a];
            const float az = wa * Watt[c]            + batt[c];
            const float ar = wa * Watt[CDIM + c]     + batt[CDIM + c];
            const float an = wa * Watt[2 * CDIM + c] + batt[2 * CDIM + c];
            const float ez = Ge[a * (3 * CDIM) + c];
            const float er = Ge[a * (3 * CDIM) + CDIM + c];
            const float en = Ge[a * (3 * CDIM) + 2 * CDIM + c];
            const float z = 1.f / (1.f + __expf(-(cz + az + ez + gd[c])));
            const float r = 1.f / (1.f + __expf(-(cr + ar + er + gd[CDIM + c])));
            const float nv = tanhf(r * cn + an + en + gd[2 * CDIM + c]);
            covN[idx] = (1.f - z) * nv + z * cov[idx];
        }
        __syncthreads();
        for (int idx = t; idx < ADM * CDIM; idx += 256) cov[idx] = covN[idx];

        // read = w @ Mve
        float rr = 0.f;
        for (int a = 0; a < ADM; ++a) rr += w[a] * Mve[a * HDIM + t];
        rd[t] = rr;
        __syncthreads();

        // gated query update
        float ag = brg[t];
        for (int i = 0; i < HDIM; ++i) ag += q[i]  * Wrg[i * HDIM + t];
        for (int i = 0; i < HDIM; ++i) ag += rd[i] * Wrg[(HDIM + i) * HDIM + t];
        const float g = 1.f / (1.f + __expf(-ag));
        tmp[t] = g * rd[t];
        __syncthreads();
        q[t] += tmp[t];
        __syncthreads();
    }

    q_out[t] = q[t];
    for (int i = t; i < ADM * CDIM; i += 256) cov_out[i] = cov[i];
}

// ---------------- coverage-gated kv attention -> context_o ----------------
__global__ __launch_bounds__(256)
void kv_context_kernel(const float* __restrict__ qv, const float* __restrict__ covv,
                       const float* __restrict__ keys,
                       const float* __restrict__ Wgk, const float* __restrict__ bgk,
                       const float* __restrict__ Mve, float* __restrict__ ctx_o)
{
    __shared__ float red[256], sc[ADM], w[ADM];
    const int t = threadIdx.x;
    const float q = qv[t];
    for (int a = 0; a < ADM; ++a) {
        float acc = bgk[t];
        for (int c = 0; c < CDIM; ++c) acc += covv[a * CDIM + c] * Wgk[c * HDIM + t];
        const float gate = 1.f / (1.f + __expf(-acc));
        red[t] = q * gate * keys[a * HDIM + t];
        __syncthreads();
        for (int s = 128; s > 0; s >>= 1) {
            if (t < s) red[t] += red[t + s];
            __syncthreads();
        }
        if (t == 0) sc[a] = red[0];
        __syncthreads();
    }
    if (t == 0) {
        float mx = sc[0];
        for (int a = 1; a < ADM; ++a) mx = fmaxf(mx, sc[a]);
        float s = 0.f;
        for (int a = 0; a < ADM; ++a) { w[a] = __expf(sc[a] - mx); s += w[a]; }
        const float inv = 1.f / s;
        for (int a = 0; a < ADM; ++a) w[a] *= inv;
    }
    __syncthreads();
    float acc = 0.f;
    for (int a = 0; a < ADM; ++a) acc += w[a] * Mve[a * HDIM + t];
    ctx_o[t] = acc;
}

// ---------------- per-row dot(q, emb[v]) : one wave32 per row ----------------
__global__ __launch_bounds__(256)
void emb_score_kernel(const float* __restrict__ qv, const float* __restrict__ emb,
                      float* __restrict__ scores)
{
    const int wv   = threadIdx.x >> 5;
    const int lane = threadIdx.x & 31;
    const int v    = blockIdx.x * 8 + wv;
    if (v >= VDIM) return;
    float acc = 0.f;
    for (int i = lane; i < HDIM; i += 32)
        acc += qv[i] * emb[(size_t)v * HDIM + i];
    for (int off = 16; off > 0; off >>= 1)
        acc += __shfl_down(acc, off, 32);
    if (lane == 0) scores[v] = acc;
}

// ---------------- softmax over V + context_e + MLP layer 1 ----------------
__global__ __launch_bounds__(256)
void final_mlp1_kernel(const float* __restrict__ qv, const float* __restrict__ ctx_o,
                       const float* __restrict__ emb, float* __restrict__ scores,
                       const float* __restrict__ Wo1, const float* __restrict__ bo1,
                       float* __restrict__ h512)
{
    __shared__ float red[256];
    __shared__ float h[3 * HDIM];
    __shared__ float mxs, sms;
    const int t = threadIdx.x;

    float m = -3.4e38f;
    for (int v = t; v < VDIM; v += 256) m = fmaxf(m, scores[v]);
    red[t] = m; __syncthreads();
    for (int s = 128; s > 0; s >>= 1) {
        if (t < s) red[t] = fmaxf(red[t], red[t + s]);
        __syncthreads();
    }
    if (t == 0) mxs = red[0];
    __syncthreads();

    float ssum = 0.f;
    for (int v = t; v < VDIM; v += 256) {
        const float p = __expf(scores[v] - mxs);
        scores[v] = p;
        ssum += p;
    }
    red[t] = ssum; __syncthreads();
    for (int s = 128; s > 0; s >>= 1) {
        if (t < s) red[t] += red[t + s];
        __syncthreads();
    }
    if (t == 0) sms = red[0];
    __syncthreads();
    const float inv = 1.f / sms;

    float acc = 0.f;
    for (int v = 0; v < VDIM; ++v) acc += scores[v] * emb[(size_t)v * HDIM + t];
    acc *= inv;

    h[t]            = fmaxf(qv[t],    0.f);
    h[HDIM + t]     = fmaxf(ctx_o[t], 0.f);
    h[2 * HDIM + t] = fmaxf(acc,      0.f);
    __syncthreads();

    for (int j = t; j < 2 * HDIM; j += 256) {
        float a2 = bo1[j];
        for (int i = 0; i < 3 * HDIM; ++i) a2 += h[i] * Wo1[i * (2 * HDIM) + j];
        h512[j] = fmaxf(a2, 0.f);
    }
}

// ---------------- output GEMV: out = h512 @ Wo2 + bo2 ----------------
__global__ __launch_bounds__(256)
void out_kernel(const float* __restrict__ h512, const float* __restrict__ Wo2,
                const float* __restrict__ bo2, float* __restrict__ out)
{
    __shared__ float hs[2 * HDIM];
    const int t = threadIdx.x;
    for (int i = t; i < 2 * HDIM; i += 256) hs[i] = h512[i];
    __syncthreads();
    const int v = blockIdx.x * 256 + t;
    if (v < VDIM) {
        float acc = bo2[v];
        for (int j = 0; j < 2 * HDIM; ++j) acc += hs[j] * Wo2[(size_t)j * VDIM + v];
        out[v] = acc;
    }
}

extern "C" void kernel_launch(void* const* d_in, const int* in_sizes, int n_in,
                              void* d_out, int out_size, void* d_ws, size_t ws_size,
                              hipStream_t stream) {
    const float* query = (const float*)d_in[0];
    const float* keys  = (const float*)d_in[1];
    const float* mvmh  = (const float*)d_in[2];
    const float* adj   = (const float*)d_in[3];
    const float* Wattn = (const float*)d_in[4];
    const float* battn = (const float*)d_in[5];
    const float* v_cov = (const float*)d_in[6];
    const float* W1    = (const float*)d_in[7];
    const float* b1    = (const float*)d_in[8];
    const float* W2    = (const float*)d_in[9];
    const float* b2    = (const float*)d_in[10];
    const float* Wgk   = (const float*)d_in[11];
    const float* bgk   = (const float*)d_in[12];
    const float* Wrg   = (const float*)d_in[13];
    const float* brg   = (const float*)d_in[14];
    const float* Wcov  = (const float*)d_in[15];
    const float* bcov  = (const float*)d_in[16];
    const float* Watt  = (const float*)d_in[17];
    const float* batt  = (const float*)d_in[18];
    const float* Wen   = (const float*)d_in[19];
    const float* ben   = (const float*)d_in[20];
    const float* Wde   = (const float*)d_in[21];
    const float* bde   = (const float*)d_in[22];
    const float* Wo1   = (const float*)d_in[23];
    const float* bo1   = (const float*)d_in[24];
    const float* Wo2   = (const float*)d_in[25];
    const float* bo2   = (const float*)d_in[26];

    float* ws     = (float*)d_ws;
    float* h1     = ws;                              // V*H
    float* t2     = h1  + (size_t)VDIM * HDIM;       // V*H
    float* emb    = t2  + (size_t)VDIM * HDIM;       // V*H
    float* Mve    = emb + (size_t)VDIM * HDIM;       // ADM*H
    float* Kw     = Mve + ADM * HDIM;                // ADM*H
    float* Ge     = Kw  + ADM * HDIM;                // ADM*3C
    float* qbuf   = Ge  + ADM * 3 * CDIM;            // H
    float* covbuf = qbuf + HDIM;                     // ADM*C
    float* ctxo   = covbuf + ADM * CDIM;             // H
    float* scores = ctxo + HDIM;                     // V
    float* h512   = scores + VDIM;                   // 2H

    const dim3 blk(256);
    auto g = [](int M, int N) { return dim3((unsigned)((N + BN - 1) / BN),
                                            (unsigned)((M + BM - 1) / BM)); };

    // GCN: h1 = relu(adj @ W1 + b1); emb = adj @ (h1 @ W2) + b2
    gemm_f32_wmma<<<g(VDIM, HDIM), blk, 0, stream>>>(adj, W1, b1,     h1,  VDIM, HDIM, VDIM, 1);
    gemm_f32_wmma<<<g(VDIM, HDIM), blk, 0, stream>>>(h1,  W2, nullptr,t2,  VDIM, HDIM, HDIM, 0);
    gemm_f32_wmma<<<g(VDIM, HDIM), blk, 0, stream>>>(adj, t2, b2,     emb, VDIM, HDIM, VDIM, 0);
    // hop-invariant precomputations
    gemm_f32_wmma<<<g(ADM, HDIM), blk, 0, stream>>>(mvmh, emb, nullptr, Mve, ADM, HDIM, VDIM, 0);
    gemm_f32_wmma<<<g(ADM, HDIM), blk, 0, stream>>>(keys, Wattn + (size_t)HDIM * HDIM, nullptr,
                                                    Kw, ADM, HDIM, HDIM, 0);
    gemm_f32_wmma<<<g(ADM, 3 * CDIM), blk, 0, stream>>>(keys, Wen, ben, Ge, ADM, 3 * CDIM, HDIM, 0);
    // sequential 20-hop scan
    hop_kernel<<<1, blk, 0, stream>>>(query, keys, Wattn, battn, v_cov, Wcov, bcov,
                                      Watt, batt, Wde, bde, Wrg, brg, Kw, Ge, Mve,
                                      qbuf, covbuf);
    // final attentions + MLP
    kv_context_kernel<<<1, blk, 0, stream>>>(qbuf, covbuf, keys, Wgk, bgk, Mve, ctxo);
    emb_score_kernel<<<dim3((VDIM + 7) / 8), blk, 0, stream>>>(qbuf, emb, scores);
    final_mlp1_kernel<<<1, blk, 0, stream>>>(qbuf, ctxo, emb, scores, Wo1, bo1, h512);
    out_kernel<<<dim3((VDIM + 255) / 256), blk, 0, stream>>>(h512, Wo2, bo2, (float*)d_out);
}